// V2LModulate_0407a_90177133347503
// MI455X (gfx1250) — compile-verified
//
#include <hip/hip_runtime.h>
#include <hip/hip_bf16.h>

// ---------------------------------------------------------------------------
// V2LModulate for MI455X (gfx1250, wave32, WMMA)
// Phase 1 (bandwidth-bound, ~537 MB): fused cosine-similarity scan.
// Phase 2 (tiny fp32 GEMM chain, M=1024): v_wmma_f32_16x16x4_f32 path.
// ---------------------------------------------------------------------------

typedef __attribute__((ext_vector_type(2))) float v2f;
typedef __attribute__((ext_vector_type(8))) float v8f;

#define B_   256
#define S_   257
#define CV_  1024
#define CT_  768
#define M_   (B_ * 4)        // 1024 rows through the MLP
#define NEG_SLOPE 0.01f
#define EPS_LN    1e-5f

// -------------------------------------------------------------------
// Kernel 1: cos_sim[b, s] = <v, i> / (||v|| * ||i||)  over CV=1024
// One wave (32 lanes) per (b, s) row; 8 float4 loads per lane/tensor.
// -------------------------------------------------------------------
__global__ __launch_bounds__(256) void cos_kernel(const float* __restrict__ vis,
                                                  const float* __restrict__ inf,
                                                  float* __restrict__ cosw) {
    const int gid  = blockIdx.x * 8 + (threadIdx.x >> 5);   // 0 .. B*256-1
    const int lane = threadIdx.x & 31;
    const int b = gid >> 8;
    const int s = gid & 255;                                 // token s+1
    const float* vp = vis + ((size_t)b * S_ + (s + 1)) * CV_;
    const float* ip = inf + ((size_t)b * S_ + (s + 1)) * CV_;

    float dot = 0.f, vv = 0.f, ii = 0.f;
#pragma unroll
    for (int i = 0; i < 8; ++i) {
        const float4 v = *(const float4*)(vp + (i * 32 + lane) * 4);
        const float4 w = *(const float4*)(ip + (i * 32 + lane) * 4);
        dot += v.x * w.x + v.y * w.y + v.z * w.z + v.w * w.w;
        vv  += v.x * v.x + v.y * v.y + v.z * v.z + v.w * v.w;
        ii  += w.x * w.x + w.y * w.y + w.z * w.z + w.w * w.w;
    }
#pragma unroll
    for (int o = 16; o > 0; o >>= 1) {
        dot += __shfl_xor(dot, o, 32);
        vv  += __shfl_xor(vv,  o, 32);
        ii  += __shfl_xor(ii,  o, 32);
    }
    if (lane == 0) cosw[gid] = dot * rsqrtf(vv) * rsqrtf(ii);
}

// -------------------------------------------------------------------
// Kernel 2: per-batch top-3 (jax tie-break: lowest index wins on ties
// via strict '>' sequential argmax) + pack X = [vis_k | inf_k].
// Rows j=0..2 are gathered tokens (idx+1), row j=3 is the CLS token 0.
// -------------------------------------------------------------------
__global__ __launch_bounds__(256) void topk_pack(const float* __restrict__ vis,
                                                 const float* __restrict__ inf,
                                                 const float* __restrict__ cosw,
                                                 float* __restrict__ X) {
    __shared__ float sc[256];
    __shared__ int   sel[3];
    const int b   = blockIdx.x;
    const int tid = threadIdx.x;
    sc[tid] = cosw[b * 256 + tid];
    __syncthreads();
    if (tid == 0) {
        for (int j = 0; j < 3; ++j) {
            float best = -__builtin_inff();
            int bi = 0;
            for (int i = 0; i < 256; ++i) {
                const float v = sc[i];
                if (v > best) { best = v; bi = i; }
            }
            sel[j] = bi;
            sc[bi] = -__builtin_inff();
        }
    }
    __syncthreads();
#pragma unroll
    for (int j = 0; j < 4; ++j) {
        const int tok = (j < 3) ? (sel[j] + 1) : 0;
        const float* vp = vis + ((size_t)b * S_ + tok) * CV_;
        const float* ip = inf + ((size_t)b * S_ + tok) * CV_;
        float* xr = X + ((size_t)(b * 4 + j)) * (2 * CV_);
        *(float4*)(xr + tid * 4)        = *(const float4*)(vp + tid * 4);
        *(float4*)(xr + CV_ + tid * 4)  = *(const float4*)(ip + tid * 4);
    }
}

// -------------------------------------------------------------------
// Kernel 3: fp32 WMMA GEMM   Y[M,N] = A[M,K] @ W[K,N] + bias
// Block = 256 threads (8 waves). Block tile 16(M) x 128(N); each wave
// owns one 16x16 D tile. K staged in LDS in chunks of 64.
// Fragment layouts per CDNA5 ISA 7.12.2:
//   A 16x4 f32 : lane m=lane&15; VGPR0/1 = K {0,1} (lanes 0-15) or {2,3} (16-31)
//   B 4x16     : VGPR0 = row K (lanes 0-15) / K+2 (16-31); VGPR1 = K+1 / K+3
//   C/D 16x16  : VGPR r = row r (lanes 0-15) / r+8 (16-31), col = lane&15
// -------------------------------------------------------------------
__global__ __launch_bounds__(256) void gemm_wmma(const float* __restrict__ A,
                                                 const float* __restrict__ W,
                                                 const float* __restrict__ bias,
                                                 float* __restrict__ Y,
                                                 int K, int N) {
    __shared__ float Alds[16 * 64];    //  4 KB
    __shared__ float Blds[64 * 128];   // 32 KB
    const int tid  = threadIdx.x;
    const int wave = tid >> 5;
    const int lane = tid & 31;
    const int m0 = blockIdx.y * 16;
    const int n0 = blockIdx.x * 128;
    const int half = lane >> 4;        // 0: K+0/K+1, 1: K+2/K+3
    const int l15  = lane & 15;

    v8f acc = {};

    for (int kb = 0; kb < K; kb += 64) {
        // Stage A chunk 16x64 (1024 f32 = 256 thr * float4)
        {
            const int idx = tid * 4;
            const int r = idx >> 6, c = idx & 63;
            *(float4*)(Alds + idx) =
                *(const float4*)(A + (size_t)(m0 + r) * K + kb + c);
        }
        // Stage B chunk 64x128 (8192 f32 = 256 thr * 8 * float4)
#pragma unroll
        for (int i = 0; i < 8; ++i) {
            const int idx = (i * 256 + tid) * 4;
            const int r = idx >> 7, c = idx & 127;
            *(float4*)(Blds + idx) =
                *(const float4*)(W + (size_t)(kb + r) * N + n0 + c);
        }
        __syncthreads();

#pragma unroll
        for (int kk = 0; kk < 64; kk += 4) {
            const int ka = kk + half * 2;
            const v2f a = *(const v2f*)(Alds + l15 * 64 + ka);   // ds_load_b64
            v2f bfrag;
            bfrag.x = Blds[(ka + 0) * 128 + wave * 16 + l15];
            bfrag.y = Blds[(ka + 1) * 128 + wave * 16 + l15];
            acc = __builtin_amdgcn_wmma_f32_16x16x4_f32(
                false, a, false, bfrag, (short)0, acc, false, false);
        }
        __syncthreads();
    }

    // Epilogue: add bias, scatter D per C/D layout.
    const int col = n0 + wave * 16 + l15;
    const float bv = bias[col];
#pragma unroll
    for (int r = 0; r < 8; ++r) {
        const int row = m0 + r + half * 8;
        Y[(size_t)row * N + col] = acc[r] + bv;
    }
}

// -------------------------------------------------------------------
// Kernel 4: fused LayerNorm (N=1024) + leaky ReLU. One block per row.
// -------------------------------------------------------------------
__global__ __launch_bounds__(256) void ln_leaky(const float* __restrict__ Xin,
                                                const float* __restrict__ g,
                                                const float* __restrict__ be,
                                                float* __restrict__ Xout) {
    __shared__ float ssum[8];
    __shared__ float ssq[8];
    const int row = blockIdx.x;
    const int tid = threadIdx.x;
    const float4 xv = *(const float4*)(Xin + (size_t)row * CV_ + tid * 4);

    float s  = xv.x + xv.y + xv.z + xv.w;
    float sq = xv.x * xv.x + xv.y * xv.y + xv.z * xv.z + xv.w * xv.w;
#pragma unroll
    for (int o = 16; o > 0; o >>= 1) {
        s  += __shfl_xor(s,  o, 32);
        sq += __shfl_xor(sq, o, 32);
    }
    if ((tid & 31) == 0) { ssum[tid >> 5] = s; ssq[tid >> 5] = sq; }
    __syncthreads();
    float ts = 0.f, tq = 0.f;
#pragma unroll
    for (int i = 0; i < 8; ++i) { ts += ssum[i]; tq += ssq[i]; }
    const float mu   = ts * (1.0f / CV_);
    const float var  = tq * (1.0f / CV_) - mu * mu;
    const float rinv = rsqrtf(var + EPS_LN);

    const float4 gv = *(const float4*)(g  + tid * 4);
    const float4 bv = *(const float4*)(be + tid * 4);
    float4 y;
    y.x = (xv.x - mu) * rinv * gv.x + bv.x;
    y.y = (xv.y - mu) * rinv * gv.y + bv.y;
    y.z = (xv.z - mu) * rinv * gv.z + bv.z;
    y.w = (xv.w - mu) * rinv * gv.w + bv.w;
    y.x = (y.x >= 0.f) ? y.x : NEG_SLOPE * y.x;
    y.y = (y.y >= 0.f) ? y.y : NEG_SLOPE * y.y;
    y.z = (y.z >= 0.f) ? y.z : NEG_SLOPE * y.z;
    y.w = (y.w >= 0.f) ? y.w : NEG_SLOPE * y.w;
    *(float4*)(Xout + (size_t)row * CV_ + tid * 4) = y;
}

// ---------------------------------------------------------------------------
extern "C" void kernel_launch(void* const* d_in, const int* in_sizes, int n_in,
                              void* d_out, int out_size, void* d_ws, size_t ws_size,
                              hipStream_t stream) {
    const float* vis = (const float*)d_in[0];
    const float* inf = (const float*)d_in[1];
    // d_in[2] = topk (==3, compile-time constant here)
    const float* W0 = (const float*)d_in[3];
    const float* b0 = (const float*)d_in[4];
    const float* g0 = (const float*)d_in[5];
    const float* be0 = (const float*)d_in[6];
    const float* W1 = (const float*)d_in[7];
    const float* b1 = (const float*)d_in[8];
    const float* g1 = (const float*)d_in[9];
    const float* be1 = (const float*)d_in[10];
    const float* W2 = (const float*)d_in[11];
    const float* b2 = (const float*)d_in[12];
    const float* g2 = (const float*)d_in[13];
    const float* be2 = (const float*)d_in[14];
    const float* W3 = (const float*)d_in[15];
    const float* b3 = (const float*)d_in[16];
    float* out = (float*)d_out;

    float* ws   = (float*)d_ws;
    float* cosw = ws;                         // B*256          = 65536
    float* X    = cosw + B_ * 256;            // M * 2CV        = 2097152
    float* bufA = X + (size_t)M_ * 2 * CV_;   // M * CV         = 1048576
    float* bufB = bufA + (size_t)M_ * CV_;    // M * CV         = 1048576

    // Phase 1: similarity scan (memory-bound, single fused pass).
    cos_kernel<<<(B_ * 256) / 8, 256, 0, stream>>>(vis, inf, cosw);
    topk_pack<<<B_, 256, 0, stream>>>(vis, inf, cosw, X);

    // Phase 2: fp32 WMMA MLP chain. Grid: (N/128, M/16).
    gemm_wmma<<<dim3(CV_ / 128, M_ / 16), 256, 0, stream>>>(X,    W0, b0, bufA, 2 * CV_, CV_);
    ln_leaky <<<M_, 256, 0, stream>>>(bufA, g0, be0, bufB);
    gemm_wmma<<<dim3(CV_ / 128, M_ / 16), 256, 0, stream>>>(bufB, W1, b1, bufA, CV_, CV_);
    ln_leaky <<<M_, 256, 0, stream>>>(bufA, g1, be1, bufB);
    gemm_wmma<<<dim3(CV_ / 128, M_ / 16), 256, 0, stream>>>(bufB, W2, b2, bufA, CV_, CV_);
    ln_leaky <<<M_, 256, 0, stream>>>(bufA, g2, be2, bufB);
    gemm_wmma<<<dim3(CT_ / 128, M_ / 16), 256, 0, stream>>>(bufB, W3, b3, out, CV_, CT_);
}